// WanTrainingModuleWithFocalLoss_72456098284071
// MI455X (gfx1250) — compile-verified
//
#include <hip/hip_runtime.h>
#include <hip/hip_bf16.h>

// ---------------------------------------------------------------------------
// WanTrainingModuleWithFocalLoss temporal loss, MI455X (gfx1250, wave32).
//
// Shapes: pred (1,81,3,480,832) f32, mask (1,81,480,832) f32, out: 1 f32.
// Strategy: group by left frame t; each block computes contributions of the
// 4 scale-pairs (t,t+1/4/8/16) for a pixel chunk -> 17-frame sliding window
// (~110MB) stays L2-resident (192MB), HBM traffic ~0.52GB (~22us @23.3TB/s).
// Wave reductions run on the matrix pipe via V_WMMA_F32_16X16X4_F32.
// ---------------------------------------------------------------------------

typedef float v2f __attribute__((ext_vector_type(2)));
typedef float v8f __attribute__((ext_vector_type(8)));

#define T_FRAMES 81
#define H_IMG 480
#define W_IMG 832
#define NPIX (H_IMG * W_IMG)      // 399360
#define NGRP (NPIX / 4)           // 99840 float4 groups per (frame,channel)
#define CH 3
#define NPAIRS 295                // 80 + 77 + 73 + 65
#define EPS2 1e-6f                // CHARBONNIER_EPS^2
#define KEEPW 0.85f               // 1 - SUBTITLE_WEIGHT
#define GRID_X 78                 // 78*256 = 19968; 99840/19968 = 5 exact iters

// Wave32-wide sum on the matrix pipe. A = {v, 0} per lane, B = all-ones
// (all-ones B makes the B VGPR layout irrelevant). Per the documented A
// layout for 16x4 f32 (lanes 0-15: K=0/1, lanes 16-31: K=2/3):
//   D[m,n] = v[lane m] + v[lane m+16]   for every n.
// D layout: lane L holds column n=L%16, VGPR r holds row m=r+8*(L>=16).
// Sum of the 8 D VGPRs + xor-16 shuffle = full 32-lane sum in every lane.
// Requires EXEC all ones (callers must be divergence-free here).
__device__ __forceinline__ float wave_sum_wmma(float v) {
    v2f a; a[0] = v;    a[1] = 0.0f;
    v2f b; b[0] = 1.0f; b[1] = 1.0f;
    v8f c = {0.f, 0.f, 0.f, 0.f, 0.f, 0.f, 0.f, 0.f};
    v8f d = __builtin_amdgcn_wmma_f32_16x16x4_f32(
        /*neg_a=*/false, a, /*neg_b=*/false, b,
        /*c_mod=*/(short)0, c, /*reuse_a=*/false, /*reuse_b=*/false);
    float s = ((d[0] + d[1]) + (d[2] + d[3])) + ((d[4] + d[5]) + (d[6] + d[7]));
    s += __shfl_xor(s, 16, 32);
    return s;
}

__global__ void wan_zero_acc(float* __restrict__ acc, int n) {
    for (int i = threadIdx.x; i < n; i += blockDim.x) acc[i] = 0.0f;
}

__global__ __launch_bounds__(256) void wan_temporal_pairs(
    const float* __restrict__ pred,   // (81,3,480,832)
    const float* __restrict__ mask,   // (81,480,832)
    float* __restrict__ acc)          // [0..294]=num, [295..589]=den(sum w)
{
    const int t = blockIdx.y;                   // left frame, 0..79
    const int scales[4] = {1, 4, 8, 16};
    const int pbase[4]  = {0, 80, 157, 230};

    float numP[4] = {0.f, 0.f, 0.f, 0.f};
    float denP[4] = {0.f, 0.f, 0.f, 0.f};

    const float4* mL  = (const float4*)(mask + (size_t)t * NPIX);
    const float4* pL0 = (const float4*)(pred + ((size_t)t * CH + 0) * NPIX);
    const float4* pL1 = (const float4*)(pred + ((size_t)t * CH + 1) * NPIX);
    const float4* pL2 = (const float4*)(pred + ((size_t)t * CH + 2) * NPIX);

    const int stride = blockDim.x * gridDim.x;  // 19968, divides NGRP exactly
    for (int g = blockIdx.x * blockDim.x + threadIdx.x; g < NGRP; g += stride) {
        const float4 m0 = mL[g];
        const float4 L0 = pL0[g];
        const float4 L1 = pL1[g];
        const float4 L2 = pL2[g];

#pragma unroll
        for (int j = 0; j < 4; ++j) {
            const int s = scales[j];
            if (t + s >= T_FRAMES) continue;    // uniform across the block
            const size_t rt = (size_t)(t + s);

            const float4 m1 = ((const float4*)(mask + rt * NPIX))[g];
            const float w0 = 1.0f - fminf(fmaxf(m0.x + m1.x, 0.0f), 1.0f) * KEEPW;
            const float w1 = 1.0f - fminf(fmaxf(m0.y + m1.y, 0.0f), 1.0f) * KEEPW;
            const float w2 = 1.0f - fminf(fmaxf(m0.z + m1.z, 0.0f), 1.0f) * KEEPW;
            const float w3 = 1.0f - fminf(fmaxf(m0.w + m1.w, 0.0f), 1.0f) * KEEPW;
            denP[j] += (w0 + w1) + (w2 + w3);

            float nacc = 0.0f;
#pragma unroll
            for (int c = 0; c < CH; ++c) {
                const float4 L = (c == 0) ? L0 : ((c == 1) ? L1 : L2);
                const float4 R = ((const float4*)(pred + (rt * CH + c) * NPIX))[g];
                const float d0 = R.x - L.x;
                const float d1 = R.y - L.y;
                const float d2 = R.z - L.z;
                const float d3 = R.w - L.w;
                nacc += __builtin_amdgcn_sqrtf(fmaf(d0, d0, EPS2)) * w0;
                nacc += __builtin_amdgcn_sqrtf(fmaf(d1, d1, EPS2)) * w1;
                nacc += __builtin_amdgcn_sqrtf(fmaf(d2, d2, EPS2)) * w2;
                nacc += __builtin_amdgcn_sqrtf(fmaf(d3, d3, EPS2)) * w3;
            }
            numP[j] += nacc;
        }
    }

    // ---- block reduction: matrix-pipe wave sums, then LDS across 8 waves ----
    __shared__ float red[8][8];                 // [wave][j*2 + {num,den}]
    const int wid  = threadIdx.x >> 5;
    const int lane = threadIdx.x & 31;
#pragma unroll
    for (int j = 0; j < 4; ++j) {
        const float ns = wave_sum_wmma(numP[j]);
        const float ds = wave_sum_wmma(denP[j]);
        if (lane == 0) { red[wid][2 * j] = ns; red[wid][2 * j + 1] = ds; }
    }
    __syncthreads();

    if (threadIdx.x == 0) {
#pragma unroll
        for (int j = 0; j < 4; ++j) {
            if (t + scales[j] >= T_FRAMES) continue;
            float n = 0.0f, d = 0.0f;
            for (int w = 0; w < 8; ++w) { n += red[w][2 * j]; d += red[w][2 * j + 1]; }
            atomicAdd(&acc[pbase[j] + t], n);
            atomicAdd(&acc[NPAIRS + pbase[j] + t], d);
        }
    }
}

__global__ void wan_finalize(const float* __restrict__ acc, float* __restrict__ out) {
    if (threadIdx.x != 0 || blockIdx.x != 0) return;
    const int scales[4] = {1, 4, 8, 16};
    const int pbase[4]  = {0, 80, 157, 230};
    float total = 0.0f;
    for (int j = 0; j < 4; ++j) {
        const int s = scales[j];
        const int npair = T_FRAMES - s;
        float sum = 0.0f;
        for (int t = 0; t < npair; ++t) {
            const float num = acc[pbase[j] + t];
            const float den = acc[NPAIRS + pbase[j] + t] * 3.0f + 1e-6f;
            sum += num / den;
        }
        total += (sum / (float)s) / (float)npair;
    }
    out[0] = 0.1f * (total / 4.0f);             // LAMBDA_TEMPORAL * avg
}

extern "C" void kernel_launch(void* const* d_in, const int* in_sizes, int n_in,
                              void* d_out, int out_size, void* d_ws, size_t ws_size,
                              hipStream_t stream) {
    (void)in_sizes; (void)n_in; (void)out_size; (void)ws_size;
    const float* pred = (const float*)d_in[0];
    const float* mask = (const float*)d_in[1];
    float* acc = (float*)d_ws;                  // 2*295 floats of scratch

    wan_zero_acc<<<1, 256, 0, stream>>>(acc, 2 * NPAIRS);
    dim3 grid(GRID_X, T_FRAMES - 1);            // 78 chunks x 80 left-frames
    wan_temporal_pairs<<<grid, 256, 0, stream>>>(pred, mask, acc);
    wan_finalize<<<1, 1, 0, stream>>>(acc, (float*)d_out);
}